// ArrdecoderModel_10256381903441
// MI455X (gfx1250) — compile-verified
//
#include <hip/hip_runtime.h>
#include <hip/hip_bf16.h>
#include <math.h>

typedef __attribute__((ext_vector_type(2))) float v2f;
typedef __attribute__((ext_vector_type(4))) float v4f;
typedef __attribute__((ext_vector_type(8))) float v8f;

#define B_ 64
#define W_ 512
#define H_ 1024
#define V_ 32000

__device__ __forceinline__ float sigmoidf_(float x) { return 1.0f / (1.0f + expf(-x)); }

// ---------------- block reduction helpers (blockDim.x multiple of 32) ----------------
__device__ __forceinline__ float blk_reduce_sum(float v, float* s) {
    int lane = threadIdx.x & 31, wv = threadIdx.x >> 5, nw = (int)(blockDim.x >> 5);
#pragma unroll
    for (int o = 16; o; o >>= 1) v += __shfl_xor(v, o, 32);
    if (lane == 0) s[wv] = v;
    __syncthreads();
    if (wv == 0) {
        v = (lane < nw) ? s[lane] : 0.0f;
#pragma unroll
        for (int o = 16; o; o >>= 1) v += __shfl_xor(v, o, 32);
        if (lane == 0) s[0] = v;
    }
    __syncthreads();
    v = s[0];
    __syncthreads();
    return v;
}

__device__ __forceinline__ float blk_reduce_max(float v, float* s) {
    int lane = threadIdx.x & 31, wv = threadIdx.x >> 5, nw = (int)(blockDim.x >> 5);
#pragma unroll
    for (int o = 16; o; o >>= 1) v = fmaxf(v, __shfl_xor(v, o, 32));
    if (lane == 0) s[wv] = v;
    __syncthreads();
    if (wv == 0) {
        v = (lane < nw) ? s[lane] : -__builtin_inff();
#pragma unroll
        for (int o = 16; o; o >>= 1) v = fmaxf(v, __shfl_xor(v, o, 32));
        if (lane == 0) s[0] = v;
    }
    __syncthreads();
    v = s[0];
    __syncthreads();
    return v;
}

// ---------------- attention + context + embedding concat ----------------
// one block (256 threads) per batch element
__global__ void attention_ctx_kernel(const float* __restrict__ enc,        // (B,W,H)
                                     const float* __restrict__ h0,         // (L,B,H)
                                     const float* __restrict__ c0,         // (L,B,H)
                                     const float* __restrict__ att_w,      // (2H,)
                                     const float* __restrict__ att_b,      // (1,)
                                     const float* __restrict__ embedding,  // (V,H)
                                     const int*   __restrict__ input,      // (B,1)
                                     float* __restrict__ xcat)             // (B,2H) = [ctx, emb]
{
    __shared__ float s_scores[W_];
    __shared__ float s_wts[W_];
    __shared__ float s_red[16];

    const int b = blockIdx.x, tid = threadIdx.x;
    const float* encb = enc + (size_t)b * W_ * H_;
    const float* hL = h0 + (size_t)1 * B_ * H_ + (size_t)b * H_;  // last layer
    const float* cL = c0 + (size_t)1 * B_ * H_ + (size_t)b * H_;

    // s_b = dot(h_last*c_last, att_w[H:]) + att_b
    float p = 0.0f;
    for (int h = tid; h < H_; h += 256) p += hL[h] * cL[h] * att_w[H_ + h];
    const float sb = blk_reduce_sum(p, s_red) + att_b[0];

    // scores[w] = dot(enc[b,w,:], att_w[:H]) + sb   (one wave per row, b128 loads)
    const int lane = tid & 31, wv = tid >> 5;
    for (int w = wv; w < W_; w += 8) {
        const v4f* row4 = (const v4f*)(encb + (size_t)w * H_);
        const v4f* aw4  = (const v4f*)att_w;
        float acc = 0.0f;
        for (int h = lane; h < H_ / 4; h += 32) {
            const v4f r = row4[h], a = aw4[h];
            acc += r.x * a.x + r.y * a.y + r.z * a.z + r.w * a.w;
        }
#pragma unroll
        for (int o = 16; o; o >>= 1) acc += __shfl_xor(acc, o, 32);
        if (lane == 0) s_scores[w] = acc + sb;
    }
    __syncthreads();

    // shifted softmax: full[0]=0, full[w]=scores[w-1]
    float m = -__builtin_inff();
    for (int w = tid; w < W_; w += 256) {
        float v = (w == 0) ? 0.0f : s_scores[w - 1];
        m = fmaxf(m, v);
    }
    m = blk_reduce_max(m, s_red);
    float se = 0.0f;
    for (int w = tid; w < W_; w += 256) {
        float v = (w == 0) ? 0.0f : s_scores[w - 1];
        se += expf(v - m);
    }
    se = blk_reduce_sum(se, s_red);
    const float inv = 1.0f / se;
    for (int w = tid; w < W_; w += 256) {
        float v = (w == 0) ? 0.0f : s_scores[w - 1];
        s_wts[w] = expf(v - m) * inv;
    }
    __syncthreads();

    // ctx[b,h] = sum_w wts[w] * enc[b,w,h]  -- one b128 per thread per row (coalesced 4KB/row)
    v4f acc = {0.f, 0.f, 0.f, 0.f};
    for (int w = 0; w < W_; ++w) {
        const float wt = s_wts[w];
        const v4f r = *(const v4f*)(encb + (size_t)w * H_ + tid * 4);
        acc.x += wt * r.x; acc.y += wt * r.y; acc.z += wt * r.z; acc.w += wt * r.w;
    }
    float* xb = xcat + (size_t)b * (2 * H_);
    *(v4f*)(xb + tid * 4) = acc;

    // embedding half of concat (b128 copy)
    const int tok = input[b];
    const float* er = embedding + (size_t)tok * H_;
    *(v4f*)(xb + H_ + tid * 4) = *(const v4f*)(er + tid * 4);
}

// ---------------- fp32 WMMA GEMM:  D[M,N] = A·W^T (+ A2·W2^T) + bias (+ bias2) ----------
// One wave owns an MT*16-row M-strip x 16-col N-tile: B fragments are loaded once and
// reused across MT accumulators (cuts streaming-weight traffic by MT).
// K stepped by 8 via one global_load_b128 per matrix per step:
//   lane half 0 loads k..k+3, half 1 loads k+4..k+7.
//   WMMA #1 uses elements {0,1}  -> K-values {k,k+1,k+4,k+5}
//   WMMA #2 uses elements {2,3}  -> K-values {k+2,k+3,k+6,k+7}
// A and B use the identical slot<->k mapping, so the K-sum is exact.
// NTW: non-temporal loads for the weight stream (single-use data, keeps L2 clean).
// C/D store layout: VGPR r -> row tm + 8*(lane>>4) + r, col tn + (lane&15).
template <int MT, bool NTW>
__global__ void wmma_gemm_nt(const float* __restrict__ A,  const float* __restrict__ Wm,
                             const float* __restrict__ A2, const float* __restrict__ Wm2,
                             const float* __restrict__ bias, const float* __restrict__ bias2,
                             float* __restrict__ D,
                             int N, int K, int K2, int tilesN, int totalWaves)
{
    const int wave = (int)((blockIdx.x * blockDim.x + threadIdx.x) >> 5);
    if (wave >= totalWaves) return;  // wave-uniform: EXEC stays all-ones for WMMA
    const int lane  = threadIdx.x & 31;
    const int tn    = (wave % tilesN) << 4;
    const int tm0   = (wave / tilesN) * (16 * MT);
    const int half  = lane >> 4;         // 0 or 1
    const int idx16 = lane & 15;
    const int ko    = half << 2;         // b128 load offset: 0 or 4 floats

    const float* brow = Wm + (size_t)(tn + idx16) * K + ko;

    v8f c[MT];
#pragma unroll
    for (int t = 0; t < MT; ++t) c[t] = (v8f){0.f, 0.f, 0.f, 0.f, 0.f, 0.f, 0.f, 0.f};

    for (int k = 0; k < K; k += 8) {
        const v4f bv = NTW ? __builtin_nontemporal_load((const v4f*)(brow + k))
                           : *(const v4f*)(brow + k);
        const v2f b0 = __builtin_shufflevector(bv, bv, 0, 1);
        const v2f b1 = __builtin_shufflevector(bv, bv, 2, 3);
#pragma unroll
        for (int t = 0; t < MT; ++t) {
            const v4f av = *(const v4f*)(A + (size_t)(tm0 + t * 16 + idx16) * K + ko + k);
            const v2f a0 = __builtin_shufflevector(av, av, 0, 1);
            const v2f a1 = __builtin_shufflevector(av, av, 2, 3);
            c[t] = __builtin_amdgcn_wmma_f32_16x16x4_f32(false, a0, false, b0, (short)0, c[t], false, false);
            c[t] = __builtin_amdgcn_wmma_f32_16x16x4_f32(false, a1, false, b1, (short)0, c[t], false, false);
        }
    }
    if (A2) {
        const float* brow2 = Wm2 + (size_t)(tn + idx16) * K2 + ko;
        for (int k = 0; k < K2; k += 8) {
            const v4f bv = NTW ? __builtin_nontemporal_load((const v4f*)(brow2 + k))
                               : *(const v4f*)(brow2 + k);
            const v2f b0 = __builtin_shufflevector(bv, bv, 0, 1);
            const v2f b1 = __builtin_shufflevector(bv, bv, 2, 3);
#pragma unroll
            for (int t = 0; t < MT; ++t) {
                const v4f av = *(const v4f*)(A2 + (size_t)(tm0 + t * 16 + idx16) * K2 + ko + k);
                const v2f a0 = __builtin_shufflevector(av, av, 0, 1);
                const v2f a1 = __builtin_shufflevector(av, av, 2, 3);
                c[t] = __builtin_amdgcn_wmma_f32_16x16x4_f32(false, a0, false, b0, (short)0, c[t], false, false);
                c[t] = __builtin_amdgcn_wmma_f32_16x16x4_f32(false, a1, false, b1, (short)0, c[t], false, false);
            }
        }
    }

    const int cn = tn + idx16;
    float bvv = bias ? bias[cn] : 0.0f;
    if (bias2) bvv += bias2[cn];
#pragma unroll
    for (int t = 0; t < MT; ++t) {
        const int cm0 = tm0 + t * 16 + (half << 3);
#pragma unroll
        for (int r = 0; r < 8; ++r) {
            D[(size_t)(cm0 + r) * N + cn] = c[t][r] + bvv;
        }
    }
}

// ---------------- LSTM gate nonlinearities ----------------
__global__ void lstm_pointwise_kernel(const float* __restrict__ gates,   // (B,4H) order i,f,g,o
                                      const float* __restrict__ c_prev,  // (B,H)
                                      float* __restrict__ h_out, float* __restrict__ c_out)
{
    const int idx = blockIdx.x * blockDim.x + threadIdx.x;
    if (idx >= B_ * H_) return;
    const int b = idx >> 10;          // /H_
    const int h = idx & (H_ - 1);
    const float* g = gates + (size_t)b * 4 * H_;
    const float ig = sigmoidf_(g[h]);
    const float fg = sigmoidf_(g[H_ + h]);
    const float gg = tanhf(g[2 * H_ + h]);
    const float og = sigmoidf_(g[3 * H_ + h]);
    const float cn = fg * c_prev[idx] + ig * gg;
    c_out[idx] = cn;
    h_out[idx] = og * tanhf(cn);
}

// ---------------- log_softmax over V per row (b128 passes) ----------------
__global__ void log_softmax_rows(const float* __restrict__ logits, float* __restrict__ out)
{
    __shared__ float s_red[16];
    const int b = blockIdx.x, tid = threadIdx.x;
    const v4f* row4 = (const v4f*)(logits + (size_t)b * V_);
    const int V4 = V_ / 4;  // 8000

    float m = -__builtin_inff();
    for (int v = tid; v < V4; v += 256) {
        const v4f x = row4[v];
        m = fmaxf(m, fmaxf(fmaxf(x.x, x.y), fmaxf(x.z, x.w)));
    }
    m = blk_reduce_max(m, s_red);
    float se = 0.0f;
    for (int v = tid; v < V4; v += 256) {
        const v4f x = row4[v];
        se += expf(x.x - m) + expf(x.y - m) + expf(x.z - m) + expf(x.w - m);
    }
    se = blk_reduce_sum(se, s_red);
    const float lse = m + logf(se);
    v4f* ob4 = (v4f*)(out + (size_t)b * V_);
    for (int v = tid; v < V4; v += 256) {
        v4f x = row4[v];
        x.x -= lse; x.y -= lse; x.z -= lse; x.w -= lse;
        ob4[v] = x;
    }
}

extern "C" void kernel_launch(void* const* d_in, const int* in_sizes, int n_in,
                              void* d_out, int out_size, void* d_ws, size_t ws_size,
                              hipStream_t stream) {
    const int*   input     = (const int*)  d_in[0];
    const float* h0        = (const float*)d_in[1];
    const float* c0        = (const float*)d_in[2];
    const float* enc       = (const float*)d_in[3];
    const float* embedding = (const float*)d_in[4];
    const float* W_ih0     = (const float*)d_in[5];
    const float* W_hh0     = (const float*)d_in[6];
    const float* b_ih0     = (const float*)d_in[7];
    const float* b_hh0     = (const float*)d_in[8];
    const float* W_ih1     = (const float*)d_in[9];
    const float* W_hh1     = (const float*)d_in[10];
    const float* b_ih1     = (const float*)d_in[11];
    const float* b_hh1     = (const float*)d_in[12];
    const float* att_w     = (const float*)d_in[13];
    const float* att_b     = (const float*)d_in[14];
    const float* input_W   = (const float*)d_in[15];
    const float* input_b   = (const float*)d_in[16];
    const float* out_W     = (const float*)d_in[17];
    const float* out_b     = (const float*)d_in[18];

    float* out = (float*)d_out;
    float* ws  = (float*)d_ws;
    float* xcat   = ws;                 // 64*2048  = 131072 floats
    float* xvec   = ws + 131072;        // 64*1024  = 65536
    float* gates  = ws + 196608;        // 64*4096  = 262144
    float* logits = ws + 458752;        // 64*32000 = 2048000

    // output tuple layout: logp (B,1,V) | h_stack (2,B,H) | c_stack (2,B,H)
    const size_t OFF_H = (size_t)B_ * V_;            // 2048000
    const size_t OFF_C = OFF_H + (size_t)2 * B_ * H_;
    float* h1 = out + OFF_H;
    float* h2 = out + OFF_H + (size_t)B_ * H_;
    float* c1 = out + OFF_C;
    float* c2 = out + OFF_C + (size_t)B_ * H_;

    // 1) attention + context + embedding concat
    attention_ctx_kernel<<<B_, 256, 0, stream>>>(enc, h0, c0, att_w, att_b, embedding, input, xcat);

    // 2) x = xcat @ input_W^T + input_b   (M=64, N=1024, K=2048), MT=1 -> 256 waves
    {
        const int tilesN = H_ / 16;
        const int total  = (B_ / 16) * tilesN;  // 256 waves
        wmma_gemm_nt<1, false><<<(total * 32 + 255) / 256, 256, 0, stream>>>(
            xcat, input_W, nullptr, nullptr, input_b, nullptr, xvec,
            H_, 2 * H_, 0, tilesN, total);
    }
    // 3) layer-0 gates = x@W_ih0^T + h0[0]@W_hh0^T + biases  (N=4096, K=1024x2), MT=2 -> 512 waves
    {
        const int tilesN = (4 * H_) / 16;
        const int total  = (B_ / 32) * tilesN;  // 512 waves
        wmma_gemm_nt<2, false><<<(total * 32 + 255) / 256, 256, 0, stream>>>(
            xvec, W_ih0, h0, W_hh0, b_ih0, b_hh0, gates,
            4 * H_, H_, H_, tilesN, total);
    }
    lstm_pointwise_kernel<<<(B_ * H_) / 256, 256, 0, stream>>>(gates, c0, h1, c1);

    // 4) layer-1 gates = h1@W_ih1^T + h0[1]@W_hh1^T + biases, MT=2 -> 512 waves
    {
        const int tilesN = (4 * H_) / 16;
        const int total  = (B_ / 32) * tilesN;
        wmma_gemm_nt<2, false><<<(total * 32 + 255) / 256, 256, 0, stream>>>(
            h1, W_ih1, h0 + (size_t)B_ * H_, W_hh1, b_ih1, b_hh1, gates,
            4 * H_, H_, H_, tilesN, total);
    }
    lstm_pointwise_kernel<<<(B_ * H_) / 256, 256, 0, stream>>>(gates, c0 + (size_t)B_ * H_, h2, c2);

    // 5) logits = h2 @ out_W^T + out_b   (M=64, N=32000, K=1024), MT=4 -> 2000 waves.
    //    out_W (128 MB) is consumed exactly once -> non-temporal loads keep it out of L2.
    {
        const int tilesN = V_ / 16;
        const int total  = (B_ / 64) * tilesN;  // 2000 waves
        wmma_gemm_nt<4, true><<<(total * 32 + 255) / 256, 256, 0, stream>>>(
            h2, out_W, nullptr, nullptr, out_b, nullptr, logits,
            V_, H_, 0, tilesN, total);
    }
    // 6) log_softmax -> logp
    log_softmax_rows<<<B_, 256, 0, stream>>>(logits, out);
}